// GATv2Regressor_7138235646499
// MI455X (gfx1250) — compile-verified
//
#include <hip/hip_runtime.h>
#include <hip/hip_bf16.h>

typedef float v2f __attribute__((ext_vector_type(2)));
typedef float v8f __attribute__((ext_vector_type(8)));

namespace {
constexpr int B   = 256;
constexpr int NPG = 512;
constexpr int N   = B * NPG;     // 131072
constexpr int KNN = 5;
constexpr int H   = 4;
constexpr int C   = 64;
constexpr int HC  = H * C;       // 256
constexpr float SLOPE = 0.2f;
}

// ---------------------------------------------------------------------------
// 1) Per-graph kNN (k=5). One 512-thread block per graph.
// ---------------------------------------------------------------------------
__global__ void knn_kernel(const float* __restrict__ pos, int* __restrict__ nbr) {
    __shared__ float px[NPG], py[NPG], pz[NPG], sq[NPG];
    const int g  = blockIdx.x;
    const int i  = threadIdx.x;          // node within graph, 0..511
    const int gi = g * NPG + i;

    const float x = pos[gi * 3 + 0];
    const float y = pos[gi * 3 + 1];
    const float z = pos[gi * 3 + 2];
    const float s = x * x + y * y + z * z;
    px[i] = x; py[i] = y; pz[i] = z; sq[i] = s;
    __syncthreads();

    // sorted ascending: bd0 <= bd1 <= ... <= bd4
    float bd0 = 1e30f, bd1 = 1e30f, bd2 = 1e30f, bd3 = 1e30f, bd4 = 1e30f;
    int   bi0 = 0,     bi1 = 0,     bi2 = 0,     bi3 = 0,     bi4 = 0;

    for (int j = 0; j < NPG; ++j) {
        if (j == i) continue;            // reference adds 1e10 on diagonal
        const float d2 = s + sq[j] - 2.0f * (x * px[j] + y * py[j] + z * pz[j]);
        if (d2 < bd4) {
            bd4 = d2; bi4 = j;
            // bubble up with strict '<' -> stable wrt ascending j (lax.top_k ties)
            if (bd4 < bd3) { float td = bd3; bd3 = bd4; bd4 = td; int ti = bi3; bi3 = bi4; bi4 = ti; }
            if (bd3 < bd2) { float td = bd2; bd2 = bd3; bd3 = td; int ti = bi2; bi2 = bi3; bi3 = ti; }
            if (bd2 < bd1) { float td = bd1; bd1 = bd2; bd2 = td; int ti = bi1; bi1 = bi2; bi2 = ti; }
            if (bd1 < bd0) { float td = bd0; bd0 = bd1; bd1 = td; int ti = bi0; bi0 = bi1; bi1 = ti; }
        }
    }
    const int base = g * NPG;
    nbr[gi * KNN + 0] = base + bi0;
    nbr[gi * KNN + 1] = base + bi1;
    nbr[gi * KNN + 2] = base + bi2;
    nbr[gi * KNN + 3] = base + bi3;
    nbr[gi * KNN + 4] = base + bi4;
}

// ---------------------------------------------------------------------------
// 2) Embedding gather: x[n, c] = emb[z[n], c]
// ---------------------------------------------------------------------------
__global__ void embed_kernel(const int* __restrict__ z, const float* __restrict__ emb,
                             float* __restrict__ x) {
    const int i = blockIdx.x * blockDim.x + threadIdx.x;  // over N*C
    const int n = i >> 6;
    const int c = i & 63;
    x[i] = emb[z[n] * C + c];
}

// ---------------------------------------------------------------------------
// 3) GEMM: Y[N,256] = X[N,64] @ W[64,256] + bias, via V_WMMA_F32_16X16X4_F32.
//    One wave computes a 16x64 output strip (4 column tiles): each A fragment
//    (global_load_b64) feeds 4 WMMAs -> X L2-read redundancy drops 16x -> 4x.
// ---------------------------------------------------------------------------
__global__ void gemm_kernel(const float* __restrict__ X, const float* __restrict__ W,
                            const float* __restrict__ bias, float* __restrict__ Y) {
    const int wave  = blockIdx.x * (blockDim.x >> 5) + (threadIdx.x >> 5);
    const int lane  = threadIdx.x & 31;
    const int mtile = wave >> 2;              // 4 column groups per row tile
    const int ngrp  = wave & 3;
    const int m0    = mtile * 16;
    const int n0    = ngrp * 64;              // 4 x 16 columns

    const int half = lane >> 4;               // 0: K pair {0,1}; 1: K pair {2,3}
    const int l16  = lane & 15;
    const int arow = m0 + l16;                // A: M = lane%16
    const int koff = half * 2;

    v8f acc0 = {}, acc1 = {}, acc2 = {}, acc3 = {};
#pragma unroll
    for (int sstep = 0; sstep < 16; ++sstep) {
        const int k0 = sstep * 4 + koff;
        // A tile 16x4 f32: v[0]=K+0, v[1]=K+1 (per half) — loaded ONCE per step
        v2f a = *(const v2f*)(X + arow * C + k0);
        // B tiles 4x16 f32 for the 4 column tiles; same K striping, N across lanes
        const float* wrow0 = W + (k0 + 0) * HC + n0 + l16;
        const float* wrow1 = W + (k0 + 1) * HC + n0 + l16;
        v2f b0, b1, b2, b3;
        b0.x = wrow0[0];  b0.y = wrow1[0];
        b1.x = wrow0[16]; b1.y = wrow1[16];
        b2.x = wrow0[32]; b2.y = wrow1[32];
        b3.x = wrow0[48]; b3.y = wrow1[48];
        acc0 = __builtin_amdgcn_wmma_f32_16x16x4_f32(false, a, false, b0, (short)0, acc0, false, false);
        acc1 = __builtin_amdgcn_wmma_f32_16x16x4_f32(false, a, false, b1, (short)0, acc1, false, false);
        acc2 = __builtin_amdgcn_wmma_f32_16x16x4_f32(false, a, false, b2, (short)0, acc2, false, false);
        acc3 = __builtin_amdgcn_wmma_f32_16x16x4_f32(false, a, false, b3, (short)0, acc3, false, false);
    }

    // C/D layout: lanes 0-15 -> M = m0 + r, N = col; lanes 16-31 -> M = m0+8+r
    const int mbase = m0 + half * 8;
#pragma unroll
    for (int t = 0; t < 4; ++t) {
        const int   col = n0 + t * 16 + l16;
        const float bv  = bias[col];
        const v8f   acc = (t == 0) ? acc0 : (t == 1) ? acc1 : (t == 2) ? acc2 : acc3;
#pragma unroll
        for (int r = 0; r < 8; ++r) {
            Y[(mbase + r) * HC + col] = acc[r] + bv;
        }
    }
}

// ---------------------------------------------------------------------------
// 4) GATv2 attention + aggregation. One wave per node; lane owns channels
//    (lane, lane+32) per head. Writes relu(head_mean + conv_bias) into Xout.
// ---------------------------------------------------------------------------
__global__ void attn_kernel(const float* __restrict__ XL, const float* __restrict__ XR,
                            const int* __restrict__ nbr, const float* __restrict__ att,
                            const float* __restrict__ cbias, float* __restrict__ Xout) {
    const int n    = blockIdx.x * (blockDim.x >> 5) + (threadIdx.x >> 5);  // node
    const int lane = threadIdx.x & 31;

    int idx[KNN];
#pragma unroll
    for (int k = 0; k < KNN; ++k) idx[k] = nbr[n * KNN + k];

    float acc0 = 0.0f, acc1 = 0.0f;
#pragma unroll
    for (int h = 0; h < H; ++h) {
        const int   base = h * C + lane;
        const float a0   = att[base];
        const float a1   = att[base + 32];
        const float xr0  = XR[n * HC + base];
        const float xr1  = XR[n * HC + base + 32];

        float e[KNN], xj0[KNN], xj1[KNN];
#pragma unroll
        for (int k = 0; k < KNN; ++k) {
            const float* p = XL + idx[k] * HC + base;
            xj0[k] = p[0];
            xj1[k] = p[32];
            float s0 = xr0 + xj0[k]; s0 = (s0 > 0.0f) ? s0 : SLOPE * s0;
            float s1 = xr1 + xj1[k]; s1 = (s1 > 0.0f) ? s1 : SLOPE * s1;
            float part = s0 * a0 + s1 * a1;
#pragma unroll
            for (int off = 16; off >= 1; off >>= 1)
                part += __shfl_xor(part, off, 32);
            e[k] = part;                 // full per-head score (all lanes)
        }
        float emax = e[0];
#pragma unroll
        for (int k = 1; k < KNN; ++k) emax = fmaxf(emax, e[k]);
        float w[KNN], esum = 0.0f;
#pragma unroll
        for (int k = 0; k < KNN; ++k) { w[k] = __expf(e[k] - emax); esum += w[k]; }
        const float inv = 1.0f / esum;
#pragma unroll
        for (int k = 0; k < KNN; ++k) {
            const float al = w[k] * inv;
            acc0 += al * xj0[k];
            acc1 += al * xj1[k];
        }
    }
    acc0 = acc0 * (1.0f / H) + cbias[lane];
    acc1 = acc1 * (1.0f / H) + cbias[lane + 32];
    Xout[n * C + lane]      = fmaxf(acc0, 0.0f);   // relu after layer
    Xout[n * C + lane + 32] = fmaxf(acc1, 0.0f);
}

// ---------------------------------------------------------------------------
// 5) Global mean pool + fc: out[g] = mean_n(x[g,n,:]) @ fc_w + fc_b
// ---------------------------------------------------------------------------
__global__ void pool_kernel(const float* __restrict__ X, const float* __restrict__ fcw,
                            const float* __restrict__ fcb, float* __restrict__ out) {
    __shared__ float red[256];
    const int g   = blockIdx.x;
    const int t   = threadIdx.x;
    const int c   = t & 63;
    const int grp = t >> 6;              // 4 node groups
    const float w = fcw[c];
    float p = 0.0f;
    for (int nn = grp; nn < NPG; nn += 4)
        p += X[(g * NPG + nn) * C + c] * w;
    red[t] = p;
    __syncthreads();
    for (int s = 128; s > 0; s >>= 1) {
        if (t < s) red[t] += red[t + s];
        __syncthreads();
    }
    if (t == 0) out[g] = red[0] * (1.0f / NPG) + fcb[0];
}

// ---------------------------------------------------------------------------
extern "C" void kernel_launch(void* const* d_in, const int* in_sizes, int n_in,
                              void* d_out, int out_size, void* d_ws, size_t ws_size,
                              hipStream_t stream) {
    const int*   z    = (const int*)  d_in[0];
    const float* pos  = (const float*)d_in[1];
    // d_in[2] = batch (unused; graphs are equal-sized)
    const float* emb  = (const float*)d_in[3];
    const float* Wl   = (const float*)d_in[4];   // [L, 64, 256]
    const float* bl   = (const float*)d_in[5];   // [L, 256]
    const float* Wr   = (const float*)d_in[6];
    const float* br   = (const float*)d_in[7];
    const float* att  = (const float*)d_in[8];   // [L, 4, 64]
    const float* cbia = (const float*)d_in[9];   // [L, 64]
    const float* fcw  = (const float*)d_in[10];  // [64]
    const float* fcb  = (const float*)d_in[11];  // [1]
    float*       out  = (float*)d_out;           // [256]

    // Workspace layout (all fully overwritten before read)
    char* ws = (char*)d_ws;
    int*   nbr = (int*)ws;                                   // N*K ints   = 2.62 MB
    float* x   = (float*)(ws + (size_t)N * KNN * 4);         // N*64 f32   = 33.5 MB
    float* xl  = (float*)(ws + (size_t)N * KNN * 4
                             + (size_t)N * C * 4);           // N*256 f32  = 134 MB
    float* xr  = xl + (size_t)N * HC;                        // N*256 f32  = 134 MB

    // 1) kNN graph
    knn_kernel<<<B, NPG, 0, stream>>>(pos, nbr);

    // 2) embedding lookup
    embed_kernel<<<(N * C) / 256, 256, 0, stream>>>(z, emb, x);

    // 3) layers
    const int gemm_blocks = (N / 16) * 4 / 8;   // 1 wave per 16x64 strip, 8 waves/block
    const int attn_blocks = N / 8;              // 1 wave per node
    for (int l = 0; l < 3; ++l) {
        const float* Wl_l = Wl + (size_t)l * C * HC;
        const float* bl_l = bl + (size_t)l * HC;
        const float* Wr_l = Wr + (size_t)l * C * HC;
        const float* br_l = br + (size_t)l * HC;
        const float* at_l = att + (size_t)l * H * C;
        const float* cb_l = cbia + (size_t)l * C;

        gemm_kernel<<<gemm_blocks, 256, 0, stream>>>(x, Wl_l, bl_l, xl);
        gemm_kernel<<<gemm_blocks, 256, 0, stream>>>(x, Wr_l, br_l, xr);
        attn_kernel<<<attn_blocks, 256, 0, stream>>>(xl, xr, nbr, at_l, cb_l, x);
    }

    // 4) pool + fc
    pool_kernel<<<B, 256, 0, stream>>>(x, fcw, fcb, out);
}